// GraphSAGE_23390391894413
// MI455X (gfx1250) — compile-verified
//
#include <hip/hip_runtime.h>
#include <math.h>

typedef float v2f __attribute__((ext_vector_type(2)));
typedef float v4f __attribute__((ext_vector_type(4)));
typedef float v8f __attribute__((ext_vector_type(8)));

#define D 128
#define LDST 132   // padded LDS row stride (floats): spreads A-frag reads across banks

// ---------------- kernel 0: zero accumulators ----------------
__global__ void sage_zero_ws(float* __restrict__ sum, float* __restrict__ cnt, int nNodes) {
    int i = blockIdx.x * blockDim.x + threadIdx.x;
    int nf4 = nNodes * (D / 4);
    if (i < nf4) {
        v4f z = {0.0f, 0.0f, 0.0f, 0.0f};
        ((v4f*)sum)[i] = z;
    }
    if (i < nNodes) cnt[i] = 0.0f;
}

// ---------------- kernel 1: scatter-add over edges ----------------
// one wave32 per edge; lane handles 4 consecutive floats of the 128-d feature
__global__ void __launch_bounds__(256)
sage_scatter(const float* __restrict__ x,
             const int* __restrict__ src,
             const int* __restrict__ dst,
             float* __restrict__ sum,
             float* __restrict__ cnt,
             int nEdges) {
    int lane = threadIdx.x & 31;
    int warp = threadIdx.x >> 5;
    int e = blockIdx.x * 8 + warp;
    if (e >= nEdges) return;
    int s = src[e];
    int d = dst[e];
    v4f v = ((const v4f*)(x + (size_t)s * D))[lane];
    float* p = sum + (size_t)d * D + lane * 4;
    atomicAdd(p + 0, v.x);
    atomicAdd(p + 1, v.y);
    atomicAdd(p + 2, v.z);
    atomicAdd(p + 3, v.w);
    if (lane == 0) atomicAdd(cnt + d, 1.0f);
}

// ---------------- kernel 2: fused dual-GEMM + bias + L2norm + relu ----------------
// block = 256 threads = 8 waves; block owns 16 nodes x 128 out-dims.
// wave w computes the 16x16 tile for out-dims [16w, 16w+16) with V_WMMA_F32_16X16X4_F32.
__global__ void __launch_bounds__(256)
sage_gemm(const float* __restrict__ x,
          const float* __restrict__ sum,
          const float* __restrict__ cnt,
          const float* __restrict__ Wl,
          const float* __restrict__ bl,
          const float* __restrict__ Wr,
          float* __restrict__ out,
          int nNodes) {
    __shared__ float meanT[16 * LDST];
    __shared__ float xT[16 * LDST];
    __shared__ float red[16];

    const int t = threadIdx.x;
    const int nodeBase = blockIdx.x * 16;

    // ---- stage mean (= sum / max(cnt,1)) and x tiles into LDS, coalesced float4 ----
    for (int j = 0; j < 2; ++j) {
        int f4   = t + 256 * j;          // 0..511 over 16x128 floats
        int row  = f4 >> 5;              // node row in tile
        int c4   = f4 & 31;              // float4 column
        int node = nodeBase + row;
        if (node >= nNodes) node = nNodes - 1;   // clamp (N divisible by 16 in practice)
        v4f sv = ((const v4f*)(sum + (size_t)node * D))[c4];
        float inv = 1.0f / fmaxf(cnt[node], 1.0f);
        sv.x *= inv; sv.y *= inv; sv.z *= inv; sv.w *= inv;
        *(v4f*)&meanT[row * LDST + c4 * 4] = sv;
        *(v4f*)&xT[row * LDST + c4 * 4] = ((const v4f*)(x + (size_t)node * D))[c4];
    }
    if (t < 16) red[t] = 0.0f;
    __syncthreads();

    const int lane = t & 31;
    const int warp = t >> 5;            // out-dim tile 0..7
    const int n    = lane & 15;         // column within tile
    const int half = lane >> 4;         // 0: K pair {0,1} / rows 0-7 ; 1: K pair {2,3} / rows 8-15
    const int colG = warp * 16 + n;     // global out-dim

    // ---- dual GEMM: D += meanTile * Wl^T  +  xTile * Wr^T ----
    v8f c = {};
    #pragma unroll
    for (int k0 = 0; k0 < D; k0 += 4) {
        int kk = k0 + 2 * half;
        v2f aM = *(const v2f*)&meanT[(lane & 15) * LDST + kk];
        v2f aX = *(const v2f*)&xT[(lane & 15) * LDST + kk];
        v2f bL = *(const v2f*)&Wl[(size_t)colG * D + kk];   // B(k,n) = Wl[n][k]
        v2f bR = *(const v2f*)&Wr[(size_t)colG * D + kk];
        c = __builtin_amdgcn_wmma_f32_16x16x4_f32(false, aM, false, bL, (short)0, c, false, false);
        c = __builtin_amdgcn_wmma_f32_16x16x4_f32(false, aX, false, bR, (short)0, c, false, false);
    }

    // ---- bias + sum-of-squares reduction ----
    // C layout: VGPR v, lane -> element (row = v + 8*half, col = n)
    float bias = bl[colG];
    float vals[8];
    #pragma unroll
    for (int v = 0; v < 8; ++v) {
        vals[v] = c[v] + bias;
        float s = vals[v] * vals[v];
        s += __shfl_xor(s, 1, 32);
        s += __shfl_xor(s, 2, 32);
        s += __shfl_xor(s, 4, 32);
        s += __shfl_xor(s, 8, 32);      // reduced over the 16 columns of this half
        if (n == 0) atomicAdd(&red[v + 8 * half], s);   // ds_add_f32, 8 waves contribute
    }
    __syncthreads();   // red complete; also: all waves done reading meanT -> safe to reuse

    // ---- normalize + relu, write tile to LDS (reuse meanT) for coalesced store ----
    float* outT = meanT;
    #pragma unroll
    for (int v = 0; v < 8; ++v) {
        int row = v + 8 * half;
        float scale = 1.0f / fmaxf(sqrtf(red[row]), 1e-12f);
        outT[row * LDST + colG] = fmaxf(vals[v] * scale, 0.0f);
    }
    __syncthreads();

    // ---- coalesced float4 stores ----
    for (int j = 0; j < 2; ++j) {
        int f4   = t + 256 * j;
        int row  = f4 >> 5;
        int c4   = f4 & 31;
        int node = nodeBase + row;
        if (node < nNodes) {
            ((v4f*)(out + (size_t)node * D))[c4] = *(const v4f*)&outT[row * LDST + c4 * 4];
        }
    }
}

extern "C" void kernel_launch(void* const* d_in, const int* in_sizes, int n_in,
                              void* d_out, int out_size, void* d_ws, size_t ws_size,
                              hipStream_t stream) {
    const float* x     = (const float*)d_in[0];
    const int*   edges = (const int*)d_in[1];
    const float* Wl    = (const float*)d_in[2];
    const float* bl    = (const float*)d_in[3];
    const float* Wr    = (const float*)d_in[4];

    int nNodes = in_sizes[0] / D;
    int nEdges = in_sizes[1] / 2;
    const int* src = edges;
    const int* dst = edges + nEdges;

    float* sum = (float*)d_ws;                    // [nNodes, 128] f32
    float* cnt = sum + (size_t)nNodes * D;        // [nNodes] f32

    float* out = (float*)d_out;

    int zItems = nNodes * (D / 4);                // float4 count dominates
    sage_zero_ws<<<(zItems + 255) / 256, 256, 0, stream>>>(sum, cnt, nNodes);
    sage_scatter<<<(nEdges + 7) / 8, 256, 0, stream>>>(x, src, dst, sum, cnt, nEdges);
    sage_gemm<<<(nNodes + 15) / 16, 256, 0, stream>>>(x, sum, cnt, Wl, bl, Wr, out, nNodes);
}